// GCN_82248623718916
// MI455X (gfx1250) — compile-verified
//
#include <hip/hip_runtime.h>

typedef float v2f __attribute__((ext_vector_type(2)));
typedef float v8f __attribute__((ext_vector_type(8)));

#define KDIM  128   // inner dimension of every GEMM (F_IN == F_HID == 128)
#define BROWS 64    // rows of A per block
#define APAD  4     // LDS padding (stride 132 floats -> bank spread)

// -----------------------------------------------------------------------------
// C[nrows x Fout] = act(A)[nrows x 128] @ Wp[128 x FP]  via V_WMMA_F32_16X16X4_F32
// FP is the compile-time padded width (multiple of 16); Wp is zero-padded so no
// load predication is needed. act = ReLU if relu_in != 0.
// 8 waves/block; wave w -> row-tile (w>>1), col tiles ct = (w&1), (w&1)+2, ...
// WMMA sits in wave-uniform control flow (EXEC all-ones).
// -----------------------------------------------------------------------------
template <int FP>
__global__ __launch_bounds__(256)
void gcn_gemm_f32(const float* __restrict__ A, const float* __restrict__ Wp,
                  float* __restrict__ Cm, int nrows, int Fout, int relu_in)
{
    __shared__ __align__(16) float sA[BROWS][KDIM + APAD];

    const int tid  = threadIdx.x;
    const int wave = tid >> 5;
    const int lane = tid & 31;
    const int row0 = blockIdx.x * BROWS;

    // Cooperative load of the 64x128 A tile (float4-coalesced), ReLU folded in.
    for (int i = tid; i < BROWS * (KDIM / 4); i += 256) {
        const int r  = i >> 5;          // KDIM/4 == 32
        const int c4 = i & 31;
        const int gr = row0 + r;
        float4 v = make_float4(0.f, 0.f, 0.f, 0.f);
        if (gr < nrows)
            v = ((const float4*)(A + (size_t)gr * KDIM))[c4];
        if (relu_in) {
            v.x = fmaxf(v.x, 0.f); v.y = fmaxf(v.y, 0.f);
            v.z = fmaxf(v.z, 0.f); v.w = fmaxf(v.w, 0.f);
        }
        *(float4*)&sA[r][c4 * 4] = v;
    }
    __syncthreads();

    const int rt = wave >> 1;             // row tile 0..3
    const int cg = wave & 1;              // column-tile phase
    const int m  = lane & 15;             // A fragment: M = lane&15
    const int kh = (lane >> 4) << 1;      // lanes 0-15: K+0/K+1, lanes 16-31: K+2/K+3
    const int mb = (lane >> 4) << 3;      // D fragment: lanes>=16 hold M=8..15
    constexpr int NCT = FP / 16;

    // Preload this wave's entire A row-strip as 32 v2f fragments (ds_load_b64,
    // 8B-aligned since kh is even). Reused across all column tiles.
    const float* sArow = &sA[rt * 16 + m][0];
    v2f afrag[KDIM / 4];
    #pragma unroll
    for (int kk = 0; kk < KDIM / 4; ++kk)
        afrag[kk] = *(const v2f*)&sArow[kk * 4 + kh];

    for (int ct = cg; ct < NCT; ct += 2) {        // wave-uniform bounds
        const int col = (ct << 4) + (lane & 15);  // B/D fragment: N = lane&15
        // One 64-bit base; all k offsets are compile-time immediates (< 64 KB).
        const float* wbase = Wp + (size_t)kh * FP + col;
        v8f acc = {0.f, 0.f, 0.f, 0.f, 0.f, 0.f, 0.f, 0.f};

        #pragma unroll
        for (int kk = 0; kk < KDIM / 4; ++kk) {
            v2f b;
            b.x = wbase[kk * 4 * FP];
            b.y = wbase[kk * 4 * FP + FP];
            acc = __builtin_amdgcn_wmma_f32_16x16x4_f32(
                      /*neg_a=*/false, afrag[kk], /*neg_b=*/false, b,
                      /*c_mod=*/(short)0, acc, /*reuse_a=*/false, /*reuse_b=*/false);
        }

        if (col < Fout) {
            #pragma unroll
            for (int i = 0; i < 8; ++i) {
                const int gr = row0 + rt * 16 + mb + i;
                if (gr < nrows)
                    Cm[(size_t)gr * Fout + col] = acc[i];
            }
        }
    }
}

// -----------------------------------------------------------------------------
// Zero-pad W[K x F] (row-major) into Wp[K x FP].
// -----------------------------------------------------------------------------
__global__ __launch_bounds__(256)
void gcn_pad_w(const float* __restrict__ W, float* __restrict__ Wp,
               int K, int F, int FP)
{
    const int i = blockIdx.x * 256 + threadIdx.x;
    if (i >= K * FP) return;
    const int r = i / FP;
    const int c = i - r * FP;
    Wp[i] = (c < F) ? W[r * F + c] : 0.f;
}

// -----------------------------------------------------------------------------
// G[n][f] = bias[f]  (float4; F % 4 == 0 so a float4 never crosses a row).
// Pre-initializes every scatter target each launch -> stateless & poison-proof.
// -----------------------------------------------------------------------------
__global__ __launch_bounds__(256)
void gcn_fill_bias4(float4* __restrict__ G, const float4* __restrict__ b4,
                    unsigned total4, unsigned F4)
{
    const unsigned gid = blockIdx.x * 256u + threadIdx.x;
    if (gid < total4)
        G[gid] = b4[gid % F4];
}

// -----------------------------------------------------------------------------
// Edge scatter-add: G[dst[e]][:] += T[src[e]][:].
// Work item = (edge, float4 chunk); CPE = F/4 is compile-time so the divide is
// shifts (CPE=32) or a magic multiply (CPE=10). float4 gather (L2-resident
// feature table) + 4x global_atomic_add_f32 (no packed-f32 atomic on CDNA5).
// -----------------------------------------------------------------------------
template <int CPE>
__global__ __launch_bounds__(256)
void gcn_scatter_add(const float* __restrict__ T, const int* __restrict__ src,
                     const int* __restrict__ dst, float* __restrict__ G,
                     unsigned nedges)
{
    constexpr int F = CPE * 4;
    const unsigned gid = blockIdx.x * 256u + threadIdx.x;
    const unsigned e   = gid / (unsigned)CPE;
    if (e >= nedges) return;
    const unsigned ch = gid - e * (unsigned)CPE;
    const int s = src[e];
    const int d = dst[e];
    const float4 v = ((const float4*)(T + (size_t)s * F))[ch];
    float* g = G + (size_t)d * F + (size_t)ch * 4;
    atomicAdd(g + 0, v.x);
    atomicAdd(g + 1, v.y);
    atomicAdd(g + 2, v.z);
    atomicAdd(g + 3, v.w);
}

// -----------------------------------------------------------------------------
extern "C" void kernel_launch(void* const* d_in, const int* in_sizes, int n_in,
                              void* d_out, int out_size, void* d_ws, size_t ws_size,
                              hipStream_t stream)
{
    const float* x  = (const float*)d_in[0];
    const int*   ei = (const int*)  d_in[1];   // [2, E] int32
    const float* W0 = (const float*)d_in[2];
    const float* b0 = (const float*)d_in[3];
    const float* W1 = (const float*)d_in[4];
    const float* b1 = (const float*)d_in[5];
    const float* W2 = (const float*)d_in[6];
    const float* b2 = (const float*)d_in[7];
    float* out = (float*)d_out;

    const int N  = in_sizes[0] / KDIM;   // 100000
    const int E  = in_sizes[1] / 2;      // 1600000
    const int FH = KDIM;                 // 128
    const int FO = in_sizes[7];          // 40
    const int FOP = 48;                  // 40 padded to 3 column tiles

    const int* src = ei;
    const int* dst = ei + E;

    float* t   = (float*)d_ws;                 // [N,128] GEMM output
    float* g   = t + (size_t)N * FH;           // [N,128] aggregation output
    float* Wp2 = g + (size_t)N * FH;           // [128,48] zero-padded W2

    const int gemm_blocks = (N + BROWS - 1) / BROWS;

    // Pad W2 once per launch (deterministic; 24 KB).
    gcn_pad_w<<<(KDIM * FOP + 255) / 256, 256, 0, stream>>>(W2, Wp2, KDIM, FO, FOP);

    // ---- Layer 0: t = x@W0 ; g = b0 ; g[dst] += t[src] ----------------------
    gcn_gemm_f32<128><<<gemm_blocks, 256, 0, stream>>>(x, W0, t, N, FH, 0);
    {
        unsigned tot4 = (unsigned)N * (FH / 4);
        gcn_fill_bias4<<<(tot4 + 255u) / 256u, 256, 0, stream>>>(
            (float4*)g, (const float4*)b0, tot4, FH / 4);
    }
    {
        unsigned tot = (unsigned)E * (FH / 4);
        gcn_scatter_add<32><<<(tot + 255u) / 256u, 256, 0, stream>>>(t, src, dst, g, (unsigned)E);
    }

    // ---- Layer 1: t = relu(g)@W1 ; g = b1 ; g[dst] += t[src] ----------------
    gcn_gemm_f32<128><<<gemm_blocks, 256, 0, stream>>>(g, W1, t, N, FH, 1);
    {
        unsigned tot4 = (unsigned)N * (FH / 4);
        gcn_fill_bias4<<<(tot4 + 255u) / 256u, 256, 0, stream>>>(
            (float4*)g, (const float4*)b1, tot4, FH / 4);
    }
    {
        unsigned tot = (unsigned)E * (FH / 4);
        gcn_scatter_add<32><<<(tot + 255u) / 256u, 256, 0, stream>>>(t, src, dst, g, (unsigned)E);
    }

    // ---- Layer 2: t = relu(g)@Wp2 (Fout=40) ; out = b2 ; out[dst] += t[src] -
    gcn_gemm_f32<48><<<gemm_blocks, 256, 0, stream>>>(g, Wp2, t, N, FO, 1);
    {
        unsigned tot4 = (unsigned)N * (FO / 4);
        gcn_fill_bias4<<<(tot4 + 255u) / 256u, 256, 0, stream>>>(
            (float4*)out, (const float4*)b2, tot4, FO / 4);
    }
    {
        unsigned tot = (unsigned)E * (FO / 4);
        gcn_scatter_add<10><<<(tot + 255u) / 256u, 256, 0, stream>>>(t, src, dst, out, (unsigned)E);
    }
}